// VecBarrelShifterLeft108_43860206027327
// MI455X (gfx1250) — compile-verified
//
#include <hip/hip_runtime.h>
#include <stdint.h>

#define DATA_BITS   108
#define SHIFT_BITS  7
#define ROWS_PER_BLOCK 64
#define THREADS     256
#define WAVES       (THREADS / 32)
#define ROWS_PER_WAVE (ROWS_PER_BLOCK / WAVES)      // 8
#define TILE_DWORDS (ROWS_PER_BLOCK * DATA_BITS)    // 6912 floats = 27648 B LDS
#define TILE_D4     (TILE_DWORDS / 4)               // 1728 float4 per full tile
#define COPY_ITERS  ((TILE_D4 + THREADS - 1) / THREADS) // 7

// Native clang vector type: accepted by __builtin_nontemporal_store and
// lowers to one 16-byte global_store_b128.
typedef float vfloat4 __attribute__((ext_vector_type(4)));

__global__ __launch_bounds__(THREADS)
void VecBarrelShifterLeft108_kernel(const float* __restrict__ X,
                                    const int*   __restrict__ shift,
                                    float*       __restrict__ out,
                                    int B)
{
    __shared__ float tile[TILE_DWORDS];

    const int tid  = threadIdx.x;
    const int lane = tid & 31;
    const int wave = tid >> 5;

    const int blockRow0 = blockIdx.x * ROWS_PER_BLOCK;
    const int tileRows  = min(ROWS_PER_BLOCK, B - blockRow0);
    const int tileD4    = (tileRows * DATA_BITS) >> 2;   // 108 = 27*4, always divisible

    // ---------------- Stage: global -> LDS via async B128 copies ----------------
    // Global side: aligned 16B transactions (row stride 432 B, 432 % 16 == 0),
    // marked non-temporal: the stream is read exactly once, so don't thrash L2.
    // LDS side: linear image of the tile (tile sits at LDS offset 0; low 32 bits
    // of the flat shared pointer are the LDS byte offset).
    const uint64_t gbase = (uint64_t)(uintptr_t)(X + (size_t)blockRow0 * DATA_BITS);
    const uint32_t lds0  = (uint32_t)(uintptr_t)(&tile[0]);

    #pragma unroll
    for (int i = 0; i < COPY_ITERS; ++i) {
        int idx4 = i * THREADS + tid;
        if (idx4 < tileD4) {
            uint32_t goff  = (uint32_t)idx4 * 16u;   // byte offset from gbase
            uint32_t laddr = lds0 + goff;            // matching LDS byte address
            asm volatile("global_load_async_to_lds_b128 %0, %1, %2 th:TH_LOAD_NT"
                         :: "v"(laddr), "v"(goff), "s"(gbase)
                         : "memory");
        }
    }
    // Each wave drains its own ASYNCcnt; the workgroup barrier then makes the
    // whole tile visible to all waves.
    asm volatile("s_wait_asynccnt 0x0" ::: "memory");
    __syncthreads();

    // ---------------- Compute: shifted gather from LDS, b128 NT stores ----------
    const int c0 = lane * 4;                         // lanes 0..26 cover 108 cols

    #pragma unroll
    for (int rr = 0; rr < ROWS_PER_WAVE; ++rr) {
        int r = wave * ROWS_PER_WAVE + rr;           // wave-uniform row in tile
        if (r < tileRows) {                          // wave-uniform guard (ballot-safe)
            int rowg = blockRow0 + r;

            // Decode 7-bit shift with one ballot: lane j holds bit of weight 2^(6-j).
            int bit = 0;
            if (lane < SHIFT_BITS)
                bit = __builtin_nontemporal_load(&shift[(size_t)rowg * SHIFT_BITS + lane]);
            unsigned m = (unsigned)__ballot(bit != 0);   // wave32: bit j = lane j
            int S = (int)(__brev(m) >> 25);              // sum bit_j << (6-j)

            if (c0 < DATA_BITS) {
                const float* trow = &tile[r * DATA_BITS];
                vfloat4 v;
                v.x = (c0 + 0 + S < DATA_BITS) ? trow[c0 + 0 + S] : 0.0f;
                v.y = (c0 + 1 + S < DATA_BITS) ? trow[c0 + 1 + S] : 0.0f;
                v.z = (c0 + 2 + S < DATA_BITS) ? trow[c0 + 2 + S] : 0.0f;
                v.w = (c0 + 3 + S < DATA_BITS) ? trow[c0 + 3 + S] : 0.0f;
                // (rowg*108 + c0)*4 bytes: rowg*432 + lane*16 -> 16B aligned.
                __builtin_nontemporal_store(
                    v, reinterpret_cast<vfloat4*>(out + (size_t)rowg * DATA_BITS + c0));
            }
        }
    }
}

extern "C" void kernel_launch(void* const* d_in, const int* in_sizes, int n_in,
                              void* d_out, int out_size, void* d_ws, size_t ws_size,
                              hipStream_t stream) {
    const float* X     = (const float*)d_in[0];
    const int*   shift = (const int*)d_in[1];
    float*       out   = (float*)d_out;

    const int B = in_sizes[0] / DATA_BITS;
    const int grid = (B + ROWS_PER_BLOCK - 1) / ROWS_PER_BLOCK;

    hipLaunchKernelGGL(VecBarrelShifterLeft108_kernel,
                       dim3(grid), dim3(THREADS), 0, stream,
                       X, shift, out, B);
}